// Mask_Predictor_GNN_56238301774547
// MI455X (gfx1250) — compile-verified
//
#include <hip/hip_runtime.h>
#include <hip/hip_bf16.h>

typedef float v2f __attribute__((ext_vector_type(2)));
typedef float v4f __attribute__((ext_vector_type(4)));
typedef float v8f __attribute__((ext_vector_type(8)));
typedef int   v4i __attribute__((ext_vector_type(4)));

#define B_ 4
#define N_ 512
#define D_ 16
#define E_ 16
#define H_ 128
#define LDS_STRIDE 20   // padded row stride (words) to avoid LDS bank conflicts

// ---- gfx1250 async global->LDS copy (guarded; falls back to sync staging) ----
#ifndef HAVE_ASYNC_LDS
#if defined(__has_builtin)
#if __has_builtin(__builtin_amdgcn_global_load_async_to_lds_b128)
#define HAVE_ASYNC_LDS 1
#endif
#endif
#endif
#ifndef HAVE_ASYNC_LDS
#define HAVE_ASYNC_LDS 0
#endif

// address-space-qualified pointer types matching the builtin's prototype:
// param0 = v4i addrspace(1)* (global), param1 = v4i addrspace(3)* (LDS)
typedef __attribute__((address_space(1))) v4i* gbl_v4i_ptr;
typedef __attribute__((address_space(3))) v4i* lds_v4i_ptr;

__device__ __forceinline__ void wait_async_then_barrier() {
#if defined(__has_builtin) && __has_builtin(__builtin_amdgcn_s_wait_asynccnt)
  __builtin_amdgcn_s_wait_asynccnt(0);
#else
  asm volatile("s_wait_asynccnt 0" ::: "memory");
#endif
  __syncthreads();
}

// ---------------------------------------------------------------------------
// K1: proj_i = node_feature @ W_edge[0:16] + b_edge ; proj_j = nf @ W_edge[16:32]
// one thread per (b,n,h)
// ---------------------------------------------------------------------------
__global__ __launch_bounds__(256) void proj_kernel(
    const float* __restrict__ nf, const float* __restrict__ W_edge,
    const float* __restrict__ b_edge,
    float* __restrict__ proj_i, float* __restrict__ proj_j) {
  int idx = blockIdx.x * 256 + threadIdx.x;       // [0, B*N*H)
  int h  = idx & (H_ - 1);
  int bn = idx >> 7;
  const float* x = nf + (long)bn * D_;
  float pi = b_edge[h];
  float pj = 0.f;
#pragma unroll
  for (int d = 0; d < D_; ++d) {
    float xv = x[d];
    pi += xv * W_edge[d * H_ + h];
    pj += xv * W_edge[(D_ + d) * H_ + h];
  }
  proj_i[idx] = pi;
  proj_j[idx] = pj;
}

// ---------------------------------------------------------------------------
// K2: per (b,i) block — edge_out[b,i,:,:] via WMMA f32 16x16x4, fused ReLU and
// adjacency-weighted message reduction.  8 waves x 16-col H-tiles, 32 j-tiles.
// ---------------------------------------------------------------------------
__global__ __launch_bounds__(256) void edge_kernel(
    const float* __restrict__ edge_feature,
    const float* __restrict__ adjacency,
    const float* __restrict__ W_edge,
    const float* __restrict__ proj_i,
    const float* __restrict__ proj_j,
    float* __restrict__ edge_out,
    float* __restrict__ message) {
  __shared__ __align__(16) float lds_ef[N_ * LDS_STRIDE];  // 40 KB
  __shared__ __align__(16) float lds_adj[N_];              // 2 KB

  const int tid = threadIdx.x;
  const int b = blockIdx.x / N_;
  const int i = blockIdx.x - b * N_;

  // ---- stage edge_feature[b,i,:,:] (512x16 f32 = 32KB) + adjacency row ----
  const v4f* ef4  = (const v4f*)(edge_feature + (((long)b * N_ + i) * (long)N_) * E_);
  const v4f* adj4 = (const v4f*)(adjacency + ((long)b * N_ + i) * N_);
#if HAVE_ASYNC_LDS
  for (int t = tid; t < (N_ * E_) / 4; t += 256) {
    int j = t >> 2;             // (t*4)/16
    int e = (t & 3) << 2;       // (t*4)%16
    __builtin_amdgcn_global_load_async_to_lds_b128(
        (gbl_v4i_ptr)(ef4 + t),
        (lds_v4i_ptr)&lds_ef[j * LDS_STRIDE + e], 0, 0);
  }
  for (int t = tid; t < N_ / 4; t += 256) {
    __builtin_amdgcn_global_load_async_to_lds_b128(
        (gbl_v4i_ptr)(adj4 + t),
        (lds_v4i_ptr)&lds_adj[t * 4], 0, 0);
  }
  wait_async_then_barrier();
#else
  for (int t = tid; t < (N_ * E_) / 4; t += 256) {
    v4f v = __builtin_nontemporal_load(ef4 + t);   // read-once stream
    int j = t >> 2;
    int e = (t & 3) << 2;
    *(v4f*)&lds_ef[j * LDS_STRIDE + e] = v;
  }
  for (int t = tid; t < N_ / 4; t += 256) {
    *(v4f*)&lds_adj[t * 4] = __builtin_nontemporal_load(adj4 + t);
  }
  __syncthreads();
#endif

  const int lane = tid & 31;
  const int wave = tid >> 5;
  const int hi = lane >> 4;     // half-wave select
  const int n  = lane & 15;     // col within tile / A-row within tile
  const int h0 = wave * 16;     // this wave's H tile

  // ---- B fragments: We = W_edge rows [2D, 2D+E), K chunks of 4 ----
  const float* We = W_edge + 2 * D_ * H_;
  v2f bfrag[4];
#pragma unroll
  for (int c = 0; c < 4; ++c) {
    int k = 4 * c + 2 * hi;
    bfrag[c].x = We[(k    ) * H_ + h0 + n];
    bfrag[c].y = We[(k + 1) * H_ + h0 + n];
  }

  const float pib = proj_i[((long)b * N_ + i) * H_ + h0 + n];

  float macc[8];
#pragma unroll
  for (int v = 0; v < 8; ++v) macc[v] = 0.f;

  float* eo_row = edge_out + (((long)b * N_ + i) * (long)N_) * H_;
  const float* pj_base = proj_j + (long)b * N_ * H_;

  for (int jt = 0; jt < N_ / 16; ++jt) {
    const int j0 = jt * 16;

    // C = proj_j[b, j, h] + (proj_i[b,i,h] + b_edge[h])
    v8f cfrag;
#pragma unroll
    for (int v = 0; v < 8; ++v) {
      int j = j0 + v + 8 * hi;
      cfrag[v] = pj_base[(long)j * H_ + h0 + n] + pib;
    }

    // D = sum_k A(16x4) x B(4x16) + C  over E=16 in four K=4 chunks
#pragma unroll
    for (int c = 0; c < 4; ++c) {
      int k = 4 * c + 2 * hi;
      v2f a = *(const v2f*)&lds_ef[(j0 + n) * LDS_STRIDE + k];
      cfrag = __builtin_amdgcn_wmma_f32_16x16x4_f32(
          false, a, false, bfrag[c], (short)0, cfrag, false, false);
    }

    // ReLU -> NT store edge_out (write-once 512MB stream: keep out of L2),
    // accumulate adjacency-weighted message in registers
#pragma unroll
    for (int v = 0; v < 8; ++v) {
      int j = j0 + v + 8 * hi;
      float d = cfrag[v];
      float r = d > 0.0f ? d : 0.0f;
      __builtin_nontemporal_store(r, &eo_row[(long)j * H_ + h0 + n]);
      macc[v] += lds_adj[j] * r;
    }
  }

  // reduce message over the 16 row-slots (8 per half-wave) -> col h0+n
  float s = 0.f;
#pragma unroll
  for (int v = 0; v < 8; ++v) s += macc[v];
  s += __shfl_xor(s, 16, 32);
  if (lane < 16) {
    message[((long)b * N_ + i) * H_ + h0 + n] = s;
  }
}

// ---------------------------------------------------------------------------
// K3: ending = relu(concat(message, node_feature) @ W_feat + b_feat)
// one thread per (b,n,h); 144-long dot product
// ---------------------------------------------------------------------------
__global__ __launch_bounds__(256) void feat_kernel(
    const float* __restrict__ message, const float* __restrict__ nf,
    const float* __restrict__ W_feat, const float* __restrict__ b_feat,
    float* __restrict__ ending) {
  int idx = blockIdx.x * 256 + threadIdx.x;
  int h  = idx & (H_ - 1);
  int bn = idx >> 7;
  const float* m = message + (long)bn * H_;
  const float* x = nf + (long)bn * D_;
  float acc = b_feat[h];
  for (int k = 0; k < H_; ++k) acc += m[k] * W_feat[k * H_ + h];
#pragma unroll
  for (int d = 0; d < D_; ++d) acc += x[d] * W_feat[(H_ + d) * H_ + h];
  ending[idx] = acc > 0.0f ? acc : 0.0f;
}

// ---------------------------------------------------------------------------
extern "C" void kernel_launch(void* const* d_in, const int* in_sizes, int n_in,
                              void* d_out, int out_size, void* d_ws, size_t ws_size,
                              hipStream_t stream) {
  const float* node_feature = (const float*)d_in[0];
  const float* adjacency    = (const float*)d_in[1];
  const float* edge_feature = (const float*)d_in[2];
  // d_in[3] = node_mask (all ones, unused by reference math)
  const float* W_edge = (const float*)d_in[4];
  const float* b_edge = (const float*)d_in[5];
  const float* W_feat = (const float*)d_in[6];
  const float* b_feat = (const float*)d_in[7];

  float* out      = (float*)d_out;
  float* ending   = out;                          // B*N*H
  float* edge_out = out + (long)B_ * N_ * H_;     // B*N*N*H

  float* ws      = (float*)d_ws;
  float* proj_i  = ws;                            // B*N*H (includes b_edge)
  float* proj_j  = ws + (long)B_ * N_ * H_;       // B*N*H
  float* message = ws + 2L * B_ * N_ * H_;        // B*N*H

  proj_kernel<<<(B_ * N_ * H_) / 256, 256, 0, stream>>>(
      node_feature, W_edge, b_edge, proj_i, proj_j);
  edge_kernel<<<B_ * N_, 256, 0, stream>>>(
      edge_feature, adjacency, W_edge, proj_i, proj_j, edge_out, message);
  feat_kernel<<<(B_ * N_ * H_) / 256, 256, 0, stream>>>(
      message, node_feature, W_feat, b_feat, ending);
}